// TTCReward_34651796144496
// MI455X (gfx1250) — compile-verified
//
#include <hip/hip_runtime.h>
#include <hip/hip_bf16.h>
#include <math.h>

// ---- problem constants (from the reference) ----
#define HORIZ     4          // H
#define T_TOTAL   34
#define TN        30         // T_TOTAL - H
#define AGENTS    128        // AGENTS_PER
#define K_TTC     1.9f       // LEAST_MIN_TTC / DT = 0.95 / 0.5

// 16-byte async DMA chunk: global(saddr + 32-bit voffset) -> LDS, ASYNCcnt-tracked.
__device__ __forceinline__ void async_copy_b128(unsigned lds_off, unsigned goff,
                                                const void* gbase) {
    asm volatile("global_load_async_to_lds_b128 %0, %1, %2"
                 :: "v"(lds_off), "v"(goff), "s"(gbase)
                 : "memory");
}

__device__ __forceinline__ void wait_async_all() {
    asm volatile("s_wait_asynccnt 0" ::: "memory");
}

__launch_bounds__(256)
__global__ void ttc_reward_kernel(const float* __restrict__ gpos,        // [N][34][2]
                                  const float* __restrict__ gyaw,        // [N][34]
                                  const float* __restrict__ gbox,        // [N][4]
                                  const unsigned char* __restrict__ gval,// [N][34] (jax bool = 1B)
                                  const int* __restrict__ ptr,           // [B+1]
                                  float* __restrict__ out)               // [B]
{
    // Per-group slabs staged via async DMA. All sizes are 16B multiples.
    __shared__ __align__(16) float         sPos[AGENTS * T_TOTAL * 2];   // 34816 B
    __shared__ __align__(16) float         sYaw[AGENTS * T_TOTAL];       // 17408 B
    __shared__ __align__(16) float         sBox[AGENTS * 4];             //  2048 B
    __shared__ __align__(16) unsigned char sVal[AGENTS * T_TOTAL];       //  4352 B
    // per-timestep ego table: {pex, pey, ce, se, ecx[4], ecy[4]}
    __shared__ __align__(16) float         sEgo[TN][12];                 //  1440 B
    __shared__ int sFlag;

    const int tid = threadIdx.x;
    const int b   = blockIdx.x;
    const int start = ptr[b];          // group start == ego global index
    const int egoLocal = 0;

    if (tid == 0) sFlag = 0;

    // ---- async stage: contiguous global slabs -> LDS in b128 chunks ----
    {
        const void* gp = (const void*)(gpos + (size_t)start * (T_TOTAL * 2));
        const unsigned lb = (unsigned)(size_t)sPos;
        #pragma unroll 1
        for (int i = tid; i < (AGENTS * T_TOTAL * 2 * 4) / 16; i += 256)
            async_copy_b128(lb + (unsigned)i * 16u, (unsigned)i * 16u, gp);
    }
    {
        const void* gp = (const void*)(gyaw + (size_t)start * T_TOTAL);
        const unsigned lb = (unsigned)(size_t)sYaw;
        #pragma unroll 1
        for (int i = tid; i < (AGENTS * T_TOTAL * 4) / 16; i += 256)
            async_copy_b128(lb + (unsigned)i * 16u, (unsigned)i * 16u, gp);
    }
    {
        const void* gp = (const void*)(gbox + (size_t)start * 4);
        const unsigned lb = (unsigned)(size_t)sBox;
        #pragma unroll 1
        for (int i = tid; i < (AGENTS * 4 * 4) / 16; i += 256)
            async_copy_b128(lb + (unsigned)i * 16u, (unsigned)i * 16u, gp);
    }
    {
        const void* gp = (const void*)(gval + (size_t)start * T_TOTAL);
        const unsigned lb = (unsigned)(size_t)sVal;
        #pragma unroll 1
        for (int i = tid; i < (AGENTS * T_TOTAL) / 16; i += 256)
            async_copy_b128(lb + (unsigned)i * 16u, (unsigned)i * 16u, gp);
    }
    wait_async_all();     // this wave's DMA has finished writing LDS
    __syncthreads();      // all waves' DMA visible to everyone

    // ---- precompute per-timestep ego pose + world corners (threads 0..29) ----
    if (tid < TN) {
        const int tg = tid + HORIZ;
        const float* pe = &sPos[((size_t)egoLocal * T_TOTAL + tg) * 2];
        const float e1x = pe[0],  e1y = pe[1];
        const float e0x = pe[-2], e0y = pe[-1];
        const float pex = fmaf(e1x - e0x, K_TTC, e1x);
        const float pey = fmaf(e1y - e0y, K_TTC, e1y);
        float se, ce; __sincosf(sYaw[(size_t)egoLocal * T_TOTAL + tg], &se, &ce);

        const float bfe  = sBox[egoLocal * 4 + 0];
        const float bre  = sBox[egoLocal * 4 + 1];
        const float ble  = sBox[egoLocal * 4 + 2];
        const float brte = sBox[egoLocal * 4 + 3];
        const float lxe[4] = {  bfe,  bfe, -bre, -bre };
        const float lye[4] = {  ble, -brte, -brte, ble };

        float* eg = sEgo[tid];
        eg[0] = pex; eg[1] = pey; eg[2] = ce; eg[3] = se;
        #pragma unroll
        for (int k = 0; k < 4; ++k) {
            eg[4 + k] = lxe[k] * ce - lye[k] * se + pex;   // ego corner x (world)
            eg[8 + k] = lxe[k] * se + lye[k] * ce + pey;   // ego corner y (world)
        }
    }
    __syncthreads();

    // ---- compute: thread = (agent j) x (half of the 30 timesteps) ----
    const int j      = tid & (AGENTS - 1);
    const int tBegin = (tid >> 7) * (TN / 2);

    const float bfe  = sBox[egoLocal * 4 + 0];
    const float bre  = sBox[egoLocal * 4 + 1];
    const float ble  = sBox[egoLocal * 4 + 2];
    const float brte = sBox[egoLocal * 4 + 3];
    const float bfj  = sBox[j * 4 + 0];
    const float brj  = sBox[j * 4 + 1];
    const float blj  = sBox[j * 4 + 2];
    const float brtj = sBox[j * 4 + 3];
    const float lxj[4] = {  bfj,  bfj, -brj, -brj };
    const float lyj[4] = {  blj, -brtj, -brtj, blj };

    bool viol = false;
    if (j != egoLocal) {
        for (int t = tBegin; t < tBegin + TN / 2; ++t) {
            const int tg = t + HORIZ;                       // global time 4..33
            if (!sVal[j * T_TOTAL + tg] || !sVal[egoLocal * T_TOTAL + tg])
                continue;

            // agent predicted pose: p1 + 1.9*(p1 - p0)
            const float* pj = &sPos[((size_t)j * T_TOTAL + tg) * 2];
            const float p1x = pj[0],  p1y = pj[1];
            const float p0x = pj[-2], p0y = pj[-1];
            const float pjx = fmaf(p1x - p0x, K_TTC, p1x);
            const float pjy = fmaf(p1y - p0y, K_TTC, p1y);
            float sj, cj; __sincosf(sYaw[(size_t)j * T_TOTAL + tg], &sj, &cj);

            // ego table for this t (broadcast LDS reads, conflict-free)
            const float* eg = sEgo[t];
            const float pex = eg[0], pey = eg[1], ce = eg[2], se = eg[3];

            float m1 = 0.0f;   // ego corners in agent frame (losses are >= 0)
            float m2 = 0.0f;   // agent corners in ego frame
            #pragma unroll
            for (int k = 0; k < 4; ++k) {
                // ego corner k (precomputed world coords)
                const float dx = eg[4 + k] - pjx, dy = eg[8 + k] - pjy;
                const float lx =  dx * cj + dy * sj;
                const float ly = -dx * sj + dy * cj;
                const float lt = fminf(fmaxf(bfj - lx, 0.f), fmaxf(brj  + lx, 0.f));
                const float la = fminf(fmaxf(blj - ly, 0.f), fmaxf(brtj + ly, 0.f));
                m1 = fmaxf(m1, fminf(lt, la));

                // agent corner k (world)
                const float ax = lxj[k] * cj - lyj[k] * sj + pjx;
                const float ay = lxj[k] * sj + lyj[k] * cj + pjy;
                const float ex = ax - pex, ey = ay - pey;
                const float qx =  ex * ce + ey * se;
                const float qy = -ex * se + ey * ce;
                const float et = fminf(fmaxf(bfe - qx, 0.f), fmaxf(bre  + qx, 0.f));
                const float ea = fminf(fmaxf(ble - qy, 0.f), fmaxf(brte + qy, 0.f));
                m2 = fmaxf(m2, fminf(et, ea));
            }
            viol |= (fmaxf(m1, m2) > 0.0f);
        }
    }

    // ---- wave32 ballot + DS atomic OR block reduction ----
    const unsigned bal = __builtin_amdgcn_ballot_w32(viol);
    if ((tid & 31) == 0 && bal != 0u)
        atomicOr(&sFlag, 1);
    __syncthreads();

    if (tid == 0)
        out[b] = sFlag ? 0.0f : 1.0f;
}

extern "C" void kernel_launch(void* const* d_in, const int* in_sizes, int n_in,
                              void* d_out, int out_size, void* d_ws, size_t ws_size,
                              hipStream_t stream) {
    const float*         pos   = (const float*)d_in[0];        // infer_position  [N,34,2]
    const float*         yaw   = (const float*)d_in[1];        // infer_heading   [N,34]
    const float*         box   = (const float*)d_in[2];        // box             [N,4]
    const unsigned char* valid = (const unsigned char*)d_in[3];// infer_valid_mask[N,34] (bool)
    // d_in[4] = batch (implied by ptr grouping; unused)
    const int*           ptr   = (const int*)d_in[5];          // ptr [B+1]
    const int B = in_sizes[5] - 1;                             // 32 egos
    float* out = (float*)d_out;

    (void)n_in; (void)out_size; (void)d_ws; (void)ws_size;
    ttc_reward_kernel<<<B, 256, 0, stream>>>(pos, yaw, box, valid, ptr, out);
}